// TransformerEncoderLayer_34617436406220
// MI455X (gfx1250) — compile-verified
//
#include <hip/hip_runtime.h>

#define T_DIM   2048
#define N_BATCH 4
#define HID     512
#define ROWS    (T_DIM * N_BATCH)   // 8192
#define FF_DIM  1024

typedef float v2f __attribute__((ext_vector_type(2)));
typedef float v8f __attribute__((ext_vector_type(8)));

// ---------------------------------------------------------------------------
// Row-wise LayerNorm (one wave32 per 512-wide row, 8 rows per block).
// Optionally adds a per-n broadcast vector (the attention out-projection) to
// the input first, and optionally writes the pre-LN sum (the residual x2).
// ---------------------------------------------------------------------------
__global__ __launch_bounds__(256) void ln_kernel(
    const float* __restrict__ X, const float* __restrict__ addv,  // addv: [4*512] or null
    const float* __restrict__ g, const float* __restrict__ b,
    float* __restrict__ Xout,     // pre-LN sum, or null
    float* __restrict__ Hout)     // normalized output
{
  const int row  = blockIdx.x * 8 + (threadIdx.x >> 5);
  const int lane = threadIdx.x & 31;
  const float* xr = X + (size_t)row * HID;
  const int n = row & (N_BATCH - 1);

  float vals[16];
  #pragma unroll
  for (int i = 0; i < 16; ++i) {
    const int c = lane + 32 * i;
    float v = xr[c];
    if (addv) v += addv[n * HID + c];
    vals[i] = v;
  }
  float s = 0.f;
  #pragma unroll
  for (int i = 0; i < 16; ++i) s += vals[i];
  #pragma unroll
  for (int off = 16; off > 0; off >>= 1) s += __shfl_xor(s, off, 32);
  const float mean = s * (1.f / HID);

  float var = 0.f;
  #pragma unroll
  for (int i = 0; i < 16; ++i) { const float d = vals[i] - mean; var += d * d; }
  #pragma unroll
  for (int off = 16; off > 0; off >>= 1) var += __shfl_xor(var, off, 32);
  const float inv = rsqrtf(var * (1.f / HID) + 1e-5f);

  #pragma unroll
  for (int i = 0; i < 16; ++i) {
    const int c = lane + 32 * i;
    if (Xout) Xout[(size_t)row * HID + c] = vals[i];
    Hout[(size_t)row * HID + c] = (vals[i] - mean) * inv * g[c] + b[c];
  }
}

// ---------------------------------------------------------------------------
// Column sum of h over t for each (n, c) — stage 1 partials (deterministic,
// no float atomics), stage 2 reduce + 1/T scale.
// ---------------------------------------------------------------------------
__global__ __launch_bounds__(256) void colsum_stage1(
    const float* __restrict__ H, float* __restrict__ partial)
{
  const int idx = blockIdx.x * 256 + threadIdx.x;  // 0..2047 == n*512 + c
  const int n = idx >> 9;
  const int c = idx & (HID - 1);
  const int t0 = blockIdx.y * 64;
  float s = 0.f;
  for (int t = t0; t < t0 + 64; ++t)
    s += H[((size_t)(t * N_BATCH + n)) * HID + c];
  partial[blockIdx.y * (N_BATCH * HID) + idx] = s;
}

__global__ __launch_bounds__(256) void colsum_stage2(
    const float* __restrict__ partial, float* __restrict__ out)
{
  const int idx = blockIdx.x * 256 + threadIdx.x;
  float s = 0.f;
  #pragma unroll
  for (int j = 0; j < 32; ++j) s += partial[j * (N_BATCH * HID) + idx];
  out[idx] = s * (1.f / T_DIM);
}

// ---------------------------------------------------------------------------
// Tiny [4,512] x [512,512] GEMV: out[n,c] = in[n,:] . W[:, coloff+c] + bias[c]
// ---------------------------------------------------------------------------
__global__ __launch_bounds__(256) void smallmm_kernel(
    const float* __restrict__ in, const float* __restrict__ W,
    int ldw, int coloff, const float* __restrict__ bias, float* __restrict__ out)
{
  __shared__ float lin[N_BATCH * HID];
  for (int i = threadIdx.x; i < N_BATCH * HID; i += 256) lin[i] = in[i];
  __syncthreads();
  const int idx = blockIdx.x * 256 + threadIdx.x;   // 2048 outputs over 8 blocks
  const int n = idx >> 9;
  const int c = idx & (HID - 1);
  float s = bias[c];
  const float* wc = W + coloff + c;
  const float* hn = &lin[n * HID];
  for (int i = 0; i < HID; ++i) s += hn[i] * wc[(size_t)i * ldw];
  out[idx] = s;
}

// ---------------------------------------------------------------------------
// FP32 WMMA GEMM: C[M,Nc] = act(A[M,K] @ B[K,Nc] + bias) (+ resid)
// Block = 8 waves, block tile 256x64. Each wave owns 32 rows x 64 cols:
// two M halves (rows m0..m0+15 and m0+16..m0+31) x four 16-col tiles,
// so every B fragment load feeds TWO v_wmma_f32_16x16x4_f32 ops
// (8 WMMA per k-step vs 10 VMEM loads).
// Fragment layouts per CDNA5 ISA 7.12.2:
//   A 16x4 : lanes 0-15 M=r hold K,K+1 ; lanes 16-31 hold K+2,K+3
//   B 4x16 : v0 = rows K / K+2 (lane halves), v1 = rows K+1 / K+3
//   C/D    : VGPR j -> row j (lanes 0-15) / j+8 (lanes 16-31), col = r
// ---------------------------------------------------------------------------
__global__ __launch_bounds__(256) void gemm_wmma_kernel(
    const float* __restrict__ A, const float* __restrict__ B,
    const float* __restrict__ bias, const float* __restrict__ resid,
    float* __restrict__ C, int Nc, int K, int ldb, int relu)
{
  const int wv   = threadIdx.x >> 5;
  const int lane = threadIdx.x & 31;
  const int hl   = lane >> 4;        // lane half
  const int r    = lane & 15;
  const int m0 = blockIdx.x * 256 + wv * 32;
  const int n0 = blockIdx.y * 64;

  v8f p0 = {}; v8f p1 = {}; v8f p2 = {}; v8f p3 = {};   // rows m0 .. m0+15
  v8f q0 = {}; v8f q1 = {}; v8f q2 = {}; v8f q3 = {};   // rows m0+16 .. m0+31

  const float* Ap0 = A + (size_t)(m0 + r) * K + 2 * hl;
  const float* Ap1 = Ap0 + (size_t)16 * K;
  const float* Bp  = B + (size_t)(2 * hl) * ldb + n0 + r;

  for (int k = 0; k < K; k += 4) {
    v2f a0, a1;
    a0.x = Ap0[k]; a0.y = Ap0[k + 1];
    a1.x = Ap1[k]; a1.y = Ap1[k + 1];
    const float* Bk = Bp + (size_t)k * ldb;
    v2f b0, b1, b2, b3;
    b0.x = Bk[0];  b0.y = Bk[ldb];
    b1.x = Bk[16]; b1.y = Bk[ldb + 16];
    b2.x = Bk[32]; b2.y = Bk[ldb + 32];
    b3.x = Bk[48]; b3.y = Bk[ldb + 48];
    p0 = __builtin_amdgcn_wmma_f32_16x16x4_f32(false, a0, false, b0, (short)0, p0, false, false);
    q0 = __builtin_amdgcn_wmma_f32_16x16x4_f32(false, a1, false, b0, (short)0, q0, false, false);
    p1 = __builtin_amdgcn_wmma_f32_16x16x4_f32(false, a0, false, b1, (short)0, p1, false, false);
    q1 = __builtin_amdgcn_wmma_f32_16x16x4_f32(false, a1, false, b1, (short)0, q1, false, false);
    p2 = __builtin_amdgcn_wmma_f32_16x16x4_f32(false, a0, false, b2, (short)0, p2, false, false);
    q2 = __builtin_amdgcn_wmma_f32_16x16x4_f32(false, a1, false, b2, (short)0, q2, false, false);
    p3 = __builtin_amdgcn_wmma_f32_16x16x4_f32(false, a0, false, b3, (short)0, p3, false, false);
    q3 = __builtin_amdgcn_wmma_f32_16x16x4_f32(false, a1, false, b3, (short)0, q3, false, false);
  }

  const float bs0 = bias[n0 + r];
  const float bs1 = bias[n0 + 16 + r];
  const float bs2 = bias[n0 + 32 + r];
  const float bs3 = bias[n0 + 48 + r];
  #pragma unroll
  for (int j = 0; j < 8; ++j) {
    // first M half
    {
      const int row = m0 + j + 8 * hl;
      const size_t base = (size_t)row * Nc + n0 + r;
      float o0 = p0[j] + bs0, o1 = p1[j] + bs1, o2 = p2[j] + bs2, o3 = p3[j] + bs3;
      if (relu) {
        o0 = fmaxf(o0, 0.f); o1 = fmaxf(o1, 0.f);
        o2 = fmaxf(o2, 0.f); o3 = fmaxf(o3, 0.f);
      }
      if (resid) {
        o0 += resid[base];      o1 += resid[base + 16];
        o2 += resid[base + 32]; o3 += resid[base + 48];
      }
      C[base]      = o0; C[base + 16] = o1;
      C[base + 32] = o2; C[base + 48] = o3;
    }
    // second M half
    {
      const int row = m0 + 16 + j + 8 * hl;
      const size_t base = (size_t)row * Nc + n0 + r;
      float o0 = q0[j] + bs0, o1 = q1[j] + bs1, o2 = q2[j] + bs2, o3 = q3[j] + bs3;
      if (relu) {
        o0 = fmaxf(o0, 0.f); o1 = fmaxf(o1, 0.f);
        o2 = fmaxf(o2, 0.f); o3 = fmaxf(o3, 0.f);
      }
      if (resid) {
        o0 += resid[base];      o1 += resid[base + 16];
        o2 += resid[base + 32]; o3 += resid[base + 48];
      }
      C[base]      = o0; C[base + 16] = o1;
      C[base + 32] = o2; C[base + 48] = o3;
    }
  }
}

// ---------------------------------------------------------------------------
extern "C" void kernel_launch(void* const* d_in, const int* in_sizes, int n_in,
                              void* d_out, int out_size, void* d_ws, size_t ws_size,
                              hipStream_t stream)
{
  const float* x     = (const float*)d_in[0];
  const float* qkv_w = (const float*)d_in[1];
  const float* qkv_b = (const float*)d_in[2];
  const float* out_w = (const float*)d_in[3];
  const float* out_b = (const float*)d_in[4];
  const float* ln1_g = (const float*)d_in[5];
  const float* ln1_b = (const float*)d_in[6];
  const float* ff1_w = (const float*)d_in[7];
  const float* ff1_b = (const float*)d_in[8];
  const float* ff2_w = (const float*)d_in[9];
  const float* ff2_b = (const float*)d_in[10];
  const float* ln2_g = (const float*)d_in[11];
  const float* ln2_b = (const float*)d_in[12];

  char* ws = (char*)d_ws;
  const size_t SZ = (size_t)ROWS * HID * sizeof(float);   // 16 MB
  float* buf0    = (float*)ws;             // h1, then reused as x2 (residual 2)
  float* buf1    = (float*)(ws + SZ);      // h2
  float* buf2    = (float*)(ws + 2 * SZ);  // ff intermediate (32 MB)
  float* partial = (float*)(ws + 4 * SZ);  // 32 x 2048 column partials
  float* hsum    = partial + 32 * (N_BATCH * HID);
  float* amean   = hsum  + N_BATCH * HID;
  float* proj    = amean + N_BATCH * HID;

  // 1) h1 = LN1(x)
  ln_kernel<<<ROWS / 8, 256, 0, stream>>>(x, nullptr, ln1_g, ln1_b, nullptr, buf0);
  // 2) hsum[n,:] = mean_t h1[t,n,:]   (SCALE==0 -> uniform attention == mean pool)
  colsum_stage1<<<dim3(8, 32), 256, 0, stream>>>(buf0, partial);
  colsum_stage2<<<8, 256, 0, stream>>>(partial, hsum);
  // 3) attn_mean = hsum @ Wv + bv   (V block of qkv at column offset 1024)
  smallmm_kernel<<<8, 256, 0, stream>>>(hsum, qkv_w, 3 * HID, 2 * HID,
                                        qkv_b + 2 * HID, amean);
  // 4) proj = attn_mean @ out_w + out_b
  smallmm_kernel<<<8, 256, 0, stream>>>(amean, out_w, HID, 0, out_b, proj);
  // 5) x2 = x + proj[n] (broadcast over t); h2 = LN2(x2)
  ln_kernel<<<ROWS / 8, 256, 0, stream>>>(x, proj, ln2_g, ln2_b, buf0, buf1);
  // 6) ffa = relu(h2 @ ff1_w + ff1_b)   [8192 x 1024]
  gemm_wmma_kernel<<<dim3(ROWS / 256, FF_DIM / 64), 256, 0, stream>>>(
      buf1, ff1_w, ff1_b, nullptr, buf2, FF_DIM, HID, FF_DIM, 1);
  // 7) out = x2 + ffa @ ff2_w + ff2_b   [8192 x 512]
  gemm_wmma_kernel<<<dim3(ROWS / 256, HID / 64), 256, 0, stream>>>(
      buf2, ff2_w, ff2_b, buf0, (float*)d_out, HID, FF_DIM, HID, 0);
}